// ManiPointSegment_44169443672114
// MI455X (gfx1250) — compile-verified
//
#include <hip/hip_runtime.h>
#include <math.h>

#define EPSV 1e-5f

typedef __attribute__((ext_vector_type(2))) float v2f;
typedef __attribute__((ext_vector_type(8))) float v8f;

// One 16x16x4 f32 WMMA step.  A: M x lda row-major (LDS), W: cout x ldw row-major
// (padded weights, so B(k,n) = W[n*ldw + k]).
// f32 A 16x4 layout: lanes 0-15 -> M, VGPR0/1 = K 0/1; lanes 16-31 -> K 2/3.
// B 4x16 symmetric.  C layout: lane&15 = N column, row = v + 8*(lane>=16).
__device__ __forceinline__ v8f wmma_f32_step(const float* A, int lda, int m0,
                                             const float* W, int ldw, int n0,
                                             int k0, int lane, v8f c) {
  v2f a, b;
  int kb = (lane & 16) ? 2 : 0;
  int m = m0 + (lane & 15);
  const float* ap = A + (size_t)m * lda + k0 + kb;
  a[0] = ap[0];
  a[1] = ap[1];
  int n = n0 + (lane & 15);
  const float* bp = W + (size_t)n * ldw + k0 + kb;
  b[0] = bp[0];
  b[1] = bp[1];
  return __builtin_amdgcn_wmma_f32_16x16x4_f32(false, a, false, b, (short)0, c,
                                               false, false);
}

// ---------------- scalar / data movement kernels ----------------

__global__ void k_mps_transpose(const float* __restrict__ in, float* __restrict__ out,
                                int Bc, int Nc) {
  int i = blockIdx.x * blockDim.x + threadIdx.x;
  int tot = Bc * Nc * 3;
  if (i >= tot) return;
  int b = i / (Nc * 3);
  int r = i % (Nc * 3);
  int n = r / 3, c = r % 3;
  out[i] = in[(size_t)b * 3 * Nc + (size_t)c * Nc + n];
}

// Farthest point sampling: one block per batch. dist[] lives in LDS.
// Tie-break on smaller index to match jnp.argmax (first max).
__global__ void k_mps_fps(const float* __restrict__ xyz, int Nc, int np,
                          int* __restrict__ idx) {
  extern __shared__ char smem[];
  float* dist = (float*)smem;
  float* rv = dist + Nc;
  int* ri = (int*)(rv + blockDim.x);
  int b = blockIdx.x, tid = threadIdx.x;
  const float* p = xyz + (size_t)b * Nc * 3;
  for (int i = tid; i < Nc; i += blockDim.x) dist[i] = 1e10f;
  __syncthreads();
  int far = 0;
  for (int j = 0; j < np; ++j) {
    if (tid == 0) idx[b * np + j] = far;
    float cx = p[far * 3 + 0], cy = p[far * 3 + 1], cz = p[far * 3 + 2];
    float bv = -1.f;
    int bi = 0;
    for (int i = tid; i < Nc; i += blockDim.x) {
      float dx = p[i * 3 + 0] - cx, dy = p[i * 3 + 1] - cy, dz = p[i * 3 + 2] - cz;
      float d = dx * dx + dy * dy + dz * dz;
      float dd = dist[i];
      if (d < dd) { dd = d; dist[i] = d; }
      if (dd > bv) { bv = dd; bi = i; }
    }
    rv[tid] = bv;
    ri[tid] = bi;
    __syncthreads();
    for (int s = blockDim.x >> 1; s > 0; s >>= 1) {
      if (tid < s) {
        float ov = rv[tid + s];
        int oi = ri[tid + s];
        if (ov > rv[tid] || (ov == rv[tid] && oi < ri[tid])) {
          rv[tid] = ov;
          ri[tid] = oi;
        }
      }
      __syncthreads();
    }
    far = ri[0];
    __syncthreads();
  }
}

__global__ void k_mps_gather3(const float* __restrict__ src, const int* __restrict__ idx,
                              float* __restrict__ dst, int Nsrc, int S, int Bc) {
  int i = blockIdx.x * blockDim.x + threadIdx.x;
  if (i >= Bc * S * 3) return;
  int c = i % 3;
  int r = i / 3;
  int b = r / S, s = r % S;
  dst[i] = src[((size_t)b * Nsrc + idx[b * S + s]) * 3 + c];
}

// Ball query: first nsample indices (ascending) within radius; pad with first.
__global__ void k_mps_ball(const float* __restrict__ xyz, int Nc,
                           const float* __restrict__ nxyz, int S, float r2, int ns,
                           int* __restrict__ out, int Bc) {
  int i = blockIdx.x * blockDim.x + threadIdx.x;
  if (i >= Bc * S) return;
  int b = i / S;
  const float* q = nxyz + (size_t)i * 3;
  float qx = q[0], qy = q[1], qz = q[2];
  const float* p = xyz + (size_t)b * Nc * 3;
  int* o = out + (size_t)i * ns;
  int cnt = 0, first = 0;
  for (int j = 0; j < Nc && cnt < ns; ++j) {
    float dx = p[j * 3 + 0] - qx, dy = p[j * 3 + 1] - qy, dz = p[j * 3 + 2] - qz;
    if (dx * dx + dy * dy + dz * dz <= r2) {
      if (cnt == 0) first = j;
      o[cnt++] = j;
    }
  }
  for (int k = cnt; k < ns; ++k) o[k] = first;
}

// Fold BN into scale/bias and zero-pad weights to cout x Kp row-major f32.
__global__ void k_mps_prep(const float* __restrict__ W, const float* __restrict__ bb,
                           const float* __restrict__ g, const float* __restrict__ bt,
                           const float* __restrict__ rm, const float* __restrict__ rv,
                           int cout, int Kin, int coutW, int Kp,
                           float* __restrict__ Wp, float* __restrict__ sc,
                           float* __restrict__ tb) {
  int i = blockIdx.x * blockDim.x + threadIdx.x;
  if (i < coutW * Kp) {
    int co = i / Kp, k = i % Kp;
    Wp[i] = (co < cout && k < Kin) ? W[co * Kin + k] : 0.f;
  }
  if (i < coutW) {
    if (g != nullptr) {
      float inv = rsqrtf(rv[i] + EPSV);
      float s0 = g[i] * inv;
      sc[i] = s0;
      tb[i] = (bb[i] - rm[i]) * s0 + bt[i];
    } else {
      sc[i] = 1.f;
      tb[i] = (i < cout && bb != nullptr) ? bb[i] : 0.f;
    }
  }
}

// ---------------- WMMA kernels ----------------

// Fused SA layer: one block per (b,s) group. Stage centered+concatenated group
// features in LDS, WMMA against padded weights, scale/bias/ReLU, max over samples.
__global__ void k_mps_sa_group(const float* __restrict__ xyz, const float* __restrict__ nxyz,
                               const float* __restrict__ pts, int cpts,
                               const int* __restrict__ ball, int Nsrc, int S, int ns,
                               int Kp, const float* __restrict__ Wp,
                               const float* __restrict__ sc, const float* __restrict__ tb,
                               int cout, float* __restrict__ out) {
  extern __shared__ char smem[];
  float* A = (float*)smem;
  int b = blockIdx.x / S, sidx = blockIdx.x % S;
  int tid = threadIdx.x;
  int K = 3 + cpts;
  const int* bl = ball + ((size_t)b * S + sidx) * ns;
  for (int i = tid; i < ns * Kp; i += blockDim.x) {
    int r = i / Kp, k = i % Kp;
    float v = 0.f;
    int si = bl[r];
    if (k < 3)
      v = xyz[((size_t)b * Nsrc + si) * 3 + k] - nxyz[((size_t)b * S + sidx) * 3 + k];
    else if (k < K)
      v = pts[((size_t)b * Nsrc + si) * cpts + (k - 3)];
    A[i] = v;
  }
  __syncthreads();
  int lane = tid & 31, wave = tid >> 5, nw = blockDim.x >> 5;
  int Mt = ns >> 4, Nt = cout >> 4;
  for (int tn = wave; tn < Nt; tn += nw) {
    int n = tn * 16 + (lane & 15);
    float s0 = sc[n], t0 = tb[n];
    float mx = 0.f;
    for (int tm = 0; tm < Mt; ++tm) {
      v8f c = {};
      for (int k0 = 0; k0 < Kp; k0 += 4)
        c = wmma_f32_step(A, Kp, tm * 16, Wp, Kp, tn * 16, k0, lane, c);
#pragma unroll
      for (int v = 0; v < 8; ++v) {
        float val = fmaxf(c[v] * s0 + t0, 0.f);
        mx = fmaxf(mx, val);
      }
    }
    mx = fmaxf(mx, __shfl_xor(mx, 16, 32));
    if (lane < 16) out[((size_t)b * S + sidx) * cout + n] = mx;
  }
}

// Generic fused GEMM + scale/bias (+ReLU): Y[m, n] over X (rows x K, stride K),
// weights coutW x K.  32 rows per block, 8 waves, one 16x16 tile per wave step.
__global__ void k_mps_gemm(const float* __restrict__ X, int rows, int K,
                           const float* __restrict__ Wp, const float* __restrict__ sc,
                           const float* __restrict__ tb, float* __restrict__ Y, int ldY,
                           int coutW, int coutOut, int relu) {
  extern __shared__ char smem[];
  float* A = (float*)smem;
  int row0 = blockIdx.x * 32;
  int tid = threadIdx.x;
  for (int i = tid; i < 32 * K; i += blockDim.x) {
    int r = i / K, k = i % K;
    int gr = row0 + r;
    A[i] = (gr < rows) ? X[(size_t)gr * K + k] : 0.f;
  }
  __syncthreads();
  int lane = tid & 31, wave = tid >> 5, nw = blockDim.x >> 5;
  int Nt = coutW >> 4;
  for (int tt = wave; tt < 2 * Nt; tt += nw) {
    int tm = tt & 1, tn = tt >> 1;
    v8f c = {};
    for (int k0 = 0; k0 < K; k0 += 4)
      c = wmma_f32_step(A, K, tm * 16, Wp, K, tn * 16, k0, lane, c);
    int n = tn * 16 + (lane & 15);
    if (n < coutOut) {
      float s0 = sc[n], t0 = tb[n];
      int mb = row0 + tm * 16 + ((lane & 16) ? 8 : 0);
#pragma unroll
      for (int v = 0; v < 8; ++v) {
        int m = mb + v;
        if (m < rows) {
          float val = c[v] * s0 + t0;
          if (relu) val = fmaxf(val, 0.f);
          Y[(size_t)m * ldY + n] = val;
        }
      }
    }
  }
}

// ---------------- misc fused glue kernels ----------------

// sa3 group-all input: rows = b*128+i, cols: [0,3)=l2xyz, [3,131)=l2p, pad to 132.
__global__ void k_mps_concat_all(const float* __restrict__ l2xyz,
                                 const float* __restrict__ l2p,
                                 float* __restrict__ xall, int Bc) {
  const int ld = 132;
  int i = blockIdx.x * blockDim.x + threadIdx.x;
  if (i >= Bc * 128 * ld) return;
  int row = i / ld, k = i % ld;
  float v = 0.f;
  if (k < 3) v = l2xyz[row * 3 + k];
  else if (k < 131) v = l2p[(size_t)row * 128 + (k - 3)];
  xall[i] = v;
}

__global__ void k_mps_rowmax(const float* __restrict__ y, int rows, int C,
                             float* __restrict__ out, int Bc) {
  int i = blockIdx.x * blockDim.x + threadIdx.x;
  if (i >= Bc * C) return;
  int b = i / C, c = i % C;
  const float* p = y + (size_t)b * rows * C + c;
  float mx = p[0];
  for (int r = 1; r < rows; ++r) mx = fmaxf(mx, p[(size_t)r * C]);
  out[i] = mx;
}

// fp3 input (S2==1 broadcast): rows = b*128+i, cols [0,128)=l2p, [128,384)=l3p[b].
__global__ void k_mps_concat_fp3(const float* __restrict__ l2p,
                                 const float* __restrict__ l3p,
                                 float* __restrict__ xb, int Bc) {
  const int ld = 384;
  int i = blockIdx.x * blockDim.x + threadIdx.x;
  if (i >= Bc * 128 * ld) return;
  int row = i / ld, k = i % ld;
  int b = row / 128;
  xb[i] = (k < 128) ? l2p[(size_t)row * 128 + k] : l3p[b * 256 + (k - 128)];
}

// 3-NN inverse-squared-distance interpolation + concat with points1 into xb.
__global__ void k_mps_interp(const float* __restrict__ xyz1, int N1,
                             const float* __restrict__ xyz2, int S2,
                             const float* __restrict__ p1, int c1,
                             const float* __restrict__ p2, int c2,
                             float* __restrict__ xb, int Bc) {
  int i = blockIdx.x * blockDim.x + threadIdx.x;
  if (i >= Bc * N1) return;
  int b = i / N1;
  const float* q = xyz1 + (size_t)i * 3;
  float qx = q[0], qy = q[1], qz = q[2];
  const float* src = xyz2 + (size_t)b * S2 * 3;
  float d1 = 1e30f, d2 = 1e30f, d3 = 1e30f;
  int i1 = 0, i2 = 0, i3 = 0;
  for (int j = 0; j < S2; ++j) {
    float dx = src[j * 3 + 0] - qx, dy = src[j * 3 + 1] - qy, dz = src[j * 3 + 2] - qz;
    float d = dx * dx + dy * dy + dz * dz;
    if (d < d1)      { d3 = d2; i3 = i2; d2 = d1; i2 = i1; d1 = d; i1 = j; }
    else if (d < d2) { d3 = d2; i3 = i2; d2 = d;  i2 = j; }
    else if (d < d3) { d3 = d;  i3 = j; }
  }
  float r1 = 1.f / (d1 + 1e-8f), r2 = 1.f / (d2 + 1e-8f), r3 = 1.f / (d3 + 1e-8f);
  float rs = r1 + r2 + r3;
  float w1 = r1 / rs, w2 = r2 / rs, w3 = r3 / rs;
  int ld = c1 + c2;
  float* row = xb + (size_t)i * ld;
  for (int c = 0; c < c1; ++c) row[c] = p1[(size_t)i * c1 + c];
  const float* a1 = p2 + ((size_t)b * S2 + i1) * c2;
  const float* a2 = p2 + ((size_t)b * S2 + i2) * c2;
  const float* a3 = p2 + ((size_t)b * S2 + i3) * c2;
  for (int c = 0; c < c2; ++c) row[c1 + c] = w1 * a1[c] + w2 * a2[c] + w3 * a3[c];
}

__global__ void k_mps_gnstats(const float* __restrict__ y, int per,
                              float* __restrict__ mu, float* __restrict__ var) {
  __shared__ float ss[256];
  __shared__ float qq[256];
  int b = blockIdx.x;
  const float* pp = y + (size_t)b * per;
  float s = 0.f, q = 0.f;
  for (int i = threadIdx.x; i < per; i += blockDim.x) {
    float v = pp[i];
    s += v;
    q += v * v;
  }
  ss[threadIdx.x] = s;
  qq[threadIdx.x] = q;
  __syncthreads();
  for (int st = blockDim.x >> 1; st > 0; st >>= 1) {
    if (threadIdx.x < st) {
      ss[threadIdx.x] += ss[threadIdx.x + st];
      qq[threadIdx.x] += qq[threadIdx.x + st];
    }
    __syncthreads();
  }
  if (threadIdx.x == 0) {
    float m = ss[0] / per;
    mu[b] = m;
    var[b] = qq[0] / per - m * m;
  }
}

__global__ void k_mps_gnapply(float* __restrict__ y, const float* __restrict__ mu,
                              const float* __restrict__ var, const float* __restrict__ g,
                              const float* __restrict__ bt, int per, int C, int Bc) {
  int i = blockIdx.x * blockDim.x + threadIdx.x;
  if (i >= Bc * per) return;
  int b = i / per, c = i % C;
  float rinv = rsqrtf(var[b] + EPSV);
  float v = (y[i] - mu[b]) * rinv * g[c] + bt[c];
  y[i] = fmaxf(v, 0.f);
}

__global__ void k_mps_logsoftmax(const float* __restrict__ z, float* __restrict__ out,
                                 int Nc, int Bc) {
  int i = blockIdx.x * blockDim.x + threadIdx.x;
  if (i >= Bc * Nc) return;
  int b = i / Nc, n = i % Nc;
  float a0 = z[(size_t)i * 2 + 0], a1 = z[(size_t)i * 2 + 1];
  float m = fmaxf(a0, a1);
  float l = m + logf(expf(a0 - m) + expf(a1 - m));
  out[(size_t)b * 2 * Nc + n] = a0 - l;
  out[(size_t)b * 2 * Nc + Nc + n] = a1 - l;
}

// ---------------- host side ----------------

extern "C" void kernel_launch(void* const* d_in, const int* in_sizes, int n_in,
                              void* d_out, int out_size, void* d_ws, size_t ws_size,
                              hipStream_t stream) {
  (void)in_sizes; (void)n_in; (void)out_size; (void)ws_size;
  constexpr int B = 16, N = 4096;
  const float* xyz_in  = (const float*)d_in[0];
  const float* goal_in = (const float*)d_in[1];
  int p = 2;
  struct BN { const float *W, *b, *g, *bt, *rm, *rv; };
  auto takeBN = [&]() {
    BN L;
    L.W  = (const float*)d_in[p++]; L.b  = (const float*)d_in[p++];
    L.g  = (const float*)d_in[p++]; L.bt = (const float*)d_in[p++];
    L.rm = (const float*)d_in[p++]; L.rv = (const float*)d_in[p++];
    return L;
  };
  BN sa1 = takeBN(), sa2 = takeBN(), sa3 = takeBN();
  BN fp3 = takeBN(), fp2 = takeBN(), f1a = takeBN(), f1b = takeBN();
  const float* c1W = (const float*)d_in[p++];
  const float* c1b = (const float*)d_in[p++];
  const float* gng = (const float*)d_in[p++];
  const float* gnb = (const float*)d_in[p++];
  const float* c2W = (const float*)d_in[p++];
  const float* c2b = (const float*)d_in[p++];

  char* base = (char*)d_ws;
  size_t pos = 0;
  auto alloc = [&](size_t nfl) {
    pos = (pos + 255) & ~(size_t)255;
    float* r = (float*)(base + pos);
    pos += nfl * 4;
    return r;
  };
  auto alloci = [&](size_t n) { return (int*)alloc(n); };

  // padded weights + folded BN scale/bias
  float *wp_sa1 = alloc(64 * 8),    *s_sa1 = alloc(64),  *t_sa1 = alloc(64);
  float *wp_sa2 = alloc(128 * 68),  *s_sa2 = alloc(128), *t_sa2 = alloc(128);
  float *wp_sa3 = alloc(256 * 132), *s_sa3 = alloc(256), *t_sa3 = alloc(256);
  float *wp_fp3 = alloc(128 * 384), *s_fp3 = alloc(128), *t_fp3 = alloc(128);
  float *wp_fp2 = alloc(64 * 192),  *s_fp2 = alloc(64),  *t_fp2 = alloc(64);
  float *wp_f1a = alloc(64 * 64),   *s_f1a = alloc(64),  *t_f1a = alloc(64);
  float *wp_f1b = alloc(64 * 64),   *s_f1b = alloc(64),  *t_f1b = alloc(64);
  float *wp_c1  = alloc(64 * 128),  *s_c1  = alloc(64),  *t_c1  = alloc(64);
  float *wp_c2  = alloc(16 * 64),   *s_c2  = alloc(16),  *t_c2  = alloc(16);

  // branch scratch (reused by both branches, which run sequentially)
  float* xyzT  = alloc((size_t)B * N * 3);
  int*   fps1  = alloci((size_t)B * 512);
  float* l1xyz = alloc((size_t)B * 512 * 3);
  int*   ball1 = alloci((size_t)B * 512 * 32);
  float* l1p   = alloc((size_t)B * 512 * 64);
  int*   fps2  = alloci((size_t)B * 128);
  float* l2xyz = alloc((size_t)B * 128 * 3);
  int*   ball2 = alloci((size_t)B * 128 * 64);
  float* l2p   = alloc((size_t)B * 128 * 128);
  float* xall  = alloc((size_t)B * 128 * 132);
  float* y3    = alloc((size_t)B * 128 * 256);
  float* l3p   = alloc((size_t)B * 256);
  float* xb3   = alloc((size_t)B * 128 * 384);
  float* l2p2  = alloc((size_t)B * 128 * 128);
  float* xb2   = alloc((size_t)B * 512 * 192);
  float* l1p2  = alloc((size_t)B * 512 * 64);
  float* xb1   = alloc((size_t)B * N * 64);
  float* y1    = alloc((size_t)B * N * 64);
  // head
  float* xc  = alloc((size_t)B * N * 128);
  float* yh  = alloc((size_t)B * N * 64);
  float* mu  = alloc(B);
  float* var = alloc(B);
  float* z   = alloc((size_t)B * N * 2);

  const int T = 256;
  auto g1 = [&](size_t n) { return dim3((unsigned)((n + T - 1) / T)); };

  // weight prep
  k_mps_prep<<<g1(64 * 8),    T, 0, stream>>>(sa1.W, sa1.b, sa1.g, sa1.bt, sa1.rm, sa1.rv, 64, 6,   64,  8,   wp_sa1, s_sa1, t_sa1);
  k_mps_prep<<<g1(128 * 68),  T, 0, stream>>>(sa2.W, sa2.b, sa2.g, sa2.bt, sa2.rm, sa2.rv, 128, 67, 128, 68,  wp_sa2, s_sa2, t_sa2);
  k_mps_prep<<<g1(256 * 132), T, 0, stream>>>(sa3.W, sa3.b, sa3.g, sa3.bt, sa3.rm, sa3.rv, 256, 131, 256, 132, wp_sa3, s_sa3, t_sa3);
  k_mps_prep<<<g1(128 * 384), T, 0, stream>>>(fp3.W, fp3.b, fp3.g, fp3.bt, fp3.rm, fp3.rv, 128, 384, 128, 384, wp_fp3, s_fp3, t_fp3);
  k_mps_prep<<<g1(64 * 192),  T, 0, stream>>>(fp2.W, fp2.b, fp2.g, fp2.bt, fp2.rm, fp2.rv, 64, 192, 64, 192, wp_fp2, s_fp2, t_fp2);
  k_mps_prep<<<g1(64 * 64),   T, 0, stream>>>(f1a.W, f1a.b, f1a.g, f1a.bt, f1a.rm, f1a.rv, 64, 64,  64, 64,  wp_f1a, s_f1a, t_f1a);
  k_mps_prep<<<g1(64 * 64),   T, 0, stream>>>(f1b.W, f1b.b, f1b.g, f1b.bt, f1b.rm, f1b.rv, 64, 64,  64, 64,  wp_f1b, s_f1b, t_f1b);
  k_mps_prep<<<g1(64 * 128),  T, 0, stream>>>(c1W, c1b, nullptr, nullptr, nullptr, nullptr, 64, 128, 64, 128, wp_c1, s_c1, t_c1);
  k_mps_prep<<<g1(16 * 64),   T, 0, stream>>>(c2W, c2b, nullptr, nullptr, nullptr, nullptr, 2, 64, 16, 64, wp_c2, s_c2, t_c2);

  auto branch = [&](const float* xin, int half) {
    k_mps_transpose<<<g1((size_t)B * N * 3), T, 0, stream>>>(xin, xyzT, B, N);
    // SA1: FPS 512, ball r=0.2 ns=32, MLP 6->64 + max
    k_mps_fps<<<B, 256, (size_t)N * 4 + 2048, stream>>>(xyzT, N, 512, fps1);
    k_mps_gather3<<<g1((size_t)B * 512 * 3), T, 0, stream>>>(xyzT, fps1, l1xyz, N, 512, B);
    k_mps_ball<<<g1((size_t)B * 512), T, 0, stream>>>(xyzT, N, l1xyz, 512, 0.04f, 32, ball1, B);
    k_mps_sa_group<<<B * 512, 256, 32 * 8 * 4, stream>>>(xyzT, l1xyz, xyzT, 3, ball1, N, 512, 32, 8, wp_sa1, s_sa1, t_sa1, 64, l1p);
    // SA2: FPS 128, ball r=0.4 ns=64, MLP 67->128 + max
    k_mps_fps<<<B, 256, 512 * 4 + 2048, stream>>>(l1xyz, 512, 128, fps2);
    k_mps_gather3<<<g1((size_t)B * 128 * 3), T, 0, stream>>>(l1xyz, fps2, l2xyz, 512, 128, B);
    k_mps_ball<<<g1((size_t)B * 128), T, 0, stream>>>(l1xyz, 512, l2xyz, 128, 0.16f, 64, ball2, B);
    k_mps_sa_group<<<B * 128, 256, 64 * 68 * 4, stream>>>(l1xyz, l2xyz, l1p, 64, ball2, 512, 128, 64, 68, wp_sa2, s_sa2, t_sa2, 128, l2p);
    // SA3 (group all): MLP 131->256 + max over 128
    k_mps_concat_all<<<g1((size_t)B * 128 * 132), T, 0, stream>>>(l2xyz, l2p, xall, B);
    k_mps_gemm<<<(B * 128) / 32, 256, 32 * 132 * 4, stream>>>(xall, B * 128, 132, wp_sa3, s_sa3, t_sa3, y3, 256, 256, 256, 1);
    k_mps_rowmax<<<g1((size_t)B * 256), T, 0, stream>>>(y3, 128, 256, l3p, B);
    // FP3 (broadcast): 384 -> 128
    k_mps_concat_fp3<<<g1((size_t)B * 128 * 384), T, 0, stream>>>(l2p, l3p, xb3, B);
    k_mps_gemm<<<(B * 128) / 32, 256, 32 * 384 * 4, stream>>>(xb3, B * 128, 384, wp_fp3, s_fp3, t_fp3, l2p2, 128, 128, 128, 1);
    // FP2: 3-NN interp 128->512, concat 64+128 -> MLP 192->64
    k_mps_interp<<<g1((size_t)B * 512), T, 0, stream>>>(l1xyz, 512, l2xyz, 128, l1p, 64, l2p2, 128, xb2, B);
    k_mps_gemm<<<(B * 512) / 32, 256, 32 * 192 * 4, stream>>>(xb2, B * 512, 192, wp_fp2, s_fp2, t_fp2, l1p2, 64, 64, 64, 1);
    // FP1: 3-NN interp 512->4096, MLP 64->64 twice; second writes into concat half
    k_mps_interp<<<g1((size_t)B * N), T, 0, stream>>>(xyzT, N, l1xyz, 512, nullptr, 0, l1p2, 64, xb1, B);
    k_mps_gemm<<<(B * N) / 32, 256, 32 * 64 * 4, stream>>>(xb1, B * N, 64, wp_f1a, s_f1a, t_f1a, y1, 64, 64, 64, 1);
    k_mps_gemm<<<(B * N) / 32, 256, 32 * 64 * 4, stream>>>(y1, B * N, 64, wp_f1b, s_f1b, t_f1b, xc + half * 64, 128, 64, 64, 1);
  };
  branch(xyz_in, 0);
  branch(goal_in, 1);

  // head: conv1 (128->64, bias, no relu) -> GroupNorm -> relu -> conv2 (64->2) -> log_softmax
  k_mps_gemm<<<(B * N) / 32, 256, 32 * 128 * 4, stream>>>(xc, B * N, 128, wp_c1, s_c1, t_c1, yh, 64, 64, 64, 0);
  k_mps_gnstats<<<B, 256, 0, stream>>>(yh, N * 64, mu, var);
  k_mps_gnapply<<<g1((size_t)B * N * 64), T, 0, stream>>>(yh, mu, var, gng, gnb, N * 64, 64, B);
  k_mps_gemm<<<(B * N) / 32, 256, 32 * 64 * 4, stream>>>(yh, B * N, 64, wp_c2, s_c2, t_c2, z, 2, 16, 2, 0);
  k_mps_logsoftmax<<<g1((size_t)B * N), T, 0, stream>>>(z, (float*)d_out, N, B);
}